// GCNConv_79774722556124
// MI455X (gfx1250) — compile-verified
//
#include <hip/hip_runtime.h>
#include <hip/hip_bf16.h>

typedef float v2f __attribute__((ext_vector_type(2)));
typedef float v4f __attribute__((ext_vector_type(4)));
typedef float v8f __attribute__((ext_vector_type(8)));

#define DIN 256
#define DOUT 64
#define WAVES_PER_BLOCK 8
#define ROWS_PER_WAVE 32

// h[row, :] = rsqrt(out_deg[row]) * (x[row, :] @ W)
// One wave computes a 32-row x 64-col tile with V_WMMA_F32_16X16X4_F32.
// W is held in LDS pair-interleaved: pair-row kp holds (W[2kp][n], W[2kp+1][n])
// contiguously at float offset kp*128 + ((2n) ^ (((kp>>1)&1)<<6)); the XOR
// swizzle puts the two lane-halves of each WMMA B-fragment on disjoint bank
// halves, and the interleave makes every B fragment a single ds_load_b64
// straight into an even-aligned VGPR pair (no repack movs).
__global__ __launch_bounds__(256) void gemm_scale_kernel(
    const float* __restrict__ x, const float* __restrict__ w,
    const int* __restrict__ colptr, float* __restrict__ h, int n) {
  __shared__ float lw[DIN * DOUT];  // 64 KB

  const int tid = threadIdx.x;
  // Cooperative load + pair-interleave of W (2048 items of 8 floats).
#pragma unroll
  for (int it = 0; it < 8; ++it) {
    const int flat = it * 256 + tid;   // 0..2047
    const int kp   = flat >> 4;        // pair-row 0..127
    const int n0   = (flat & 15) * 4;  // col quad
    v4f g0 = *(const v4f*)(w + (size_t)(2 * kp) * DOUT + n0);
    v4f g1 = *(const v4f*)(w + (size_t)(2 * kp + 1) * DOUT + n0);
    const int sw = ((kp >> 1) & 1) << 6;
    float* dst = lw + kp * 128 + ((n0 * 2) ^ sw);
    v4f p0 = {g0.x, g1.x, g0.y, g1.y};
    v4f p1 = {g0.z, g1.z, g0.w, g1.w};
    *(v4f*)(dst)     = p0;
    *(v4f*)(dst + 4) = p1;
  }
  __syncthreads();

  const int wave = tid >> 5;
  const int lane = tid & 31;
  const int half = lane >> 4;  // 0: lanes 0-15, 1: lanes 16-31
  const int t    = lane & 15;

  const int row0 = (blockIdx.x * WAVES_PER_BLOCK + wave) * ROWS_PER_WAVE;
  if (row0 >= n) return;  // wave-uniform: EXEC all-1s for WMMA

  v8f ae0 = {}, ae1 = {}, ae2 = {}, ae3 = {};  // rows row0..row0+15
  v8f af0 = {}, af1 = {}, af2 = {}, af3 = {};  // rows row0+16..row0+31

  const float* xr0 = x + (size_t)(row0 + t) * DIN;
  const float* xr1 = xr0 + (size_t)16 * DIN;
  // K permutation: iteration u covers WMMA steps 2u and 2u+1.
  //  step 2u  : half h uses cols (8u+4h, 8u+4h+1)  -> pair-row kpe = 4u+2h
  //  step 2u+1: half h uses cols (8u+4h+2, 8u+4h+3)-> pair-row kpo = kpe+1
  // For both, ((kp>>1)&1) == h, matching the store-side swizzle.
  const float* bbase = lw + ((2 * t) ^ (half << 6));
  for (int u = 0; u < DIN / 8; ++u) {
    const int cbase = u * 8 + 4 * half;
    v4f a0 = *(const v4f*)(xr0 + cbase);  // steps 2u (.xy) and 2u+1 (.zw)
    v4f a1 = *(const v4f*)(xr1 + cbase);

    const float* be = bbase + (4 * u + 2 * half) * 128;
    const float* bo = be + 128;
    v2f be0 = *(const v2f*)(be);       v2f bo0 = *(const v2f*)(bo);
    v2f be1 = *(const v2f*)(be + 32);  v2f bo1 = *(const v2f*)(bo + 32);
    v2f be2 = *(const v2f*)(be + 64);  v2f bo2 = *(const v2f*)(bo + 64);
    v2f be3 = *(const v2f*)(be + 96);  v2f bo3 = *(const v2f*)(bo + 96);

    v2f a0e = a0.xy, a0o = a0.zw, a1e = a1.xy, a1o = a1.zw;

    ae0 = __builtin_amdgcn_wmma_f32_16x16x4_f32(false, a0e, false, be0, (short)0, ae0, false, false);
    ae1 = __builtin_amdgcn_wmma_f32_16x16x4_f32(false, a0e, false, be1, (short)0, ae1, false, false);
    ae2 = __builtin_amdgcn_wmma_f32_16x16x4_f32(false, a0e, false, be2, (short)0, ae2, false, false);
    ae3 = __builtin_amdgcn_wmma_f32_16x16x4_f32(false, a0e, false, be3, (short)0, ae3, false, false);
    af0 = __builtin_amdgcn_wmma_f32_16x16x4_f32(false, a1e, false, be0, (short)0, af0, false, false);
    af1 = __builtin_amdgcn_wmma_f32_16x16x4_f32(false, a1e, false, be1, (short)0, af1, false, false);
    af2 = __builtin_amdgcn_wmma_f32_16x16x4_f32(false, a1e, false, be2, (short)0, af2, false, false);
    af3 = __builtin_amdgcn_wmma_f32_16x16x4_f32(false, a1e, false, be3, (short)0, af3, false, false);
    ae0 = __builtin_amdgcn_wmma_f32_16x16x4_f32(false, a0o, false, bo0, (short)0, ae0, false, false);
    ae1 = __builtin_amdgcn_wmma_f32_16x16x4_f32(false, a0o, false, bo1, (short)0, ae1, false, false);
    ae2 = __builtin_amdgcn_wmma_f32_16x16x4_f32(false, a0o, false, bo2, (short)0, ae2, false, false);
    ae3 = __builtin_amdgcn_wmma_f32_16x16x4_f32(false, a0o, false, bo3, (short)0, ae3, false, false);
    af0 = __builtin_amdgcn_wmma_f32_16x16x4_f32(false, a1o, false, bo0, (short)0, af0, false, false);
    af1 = __builtin_amdgcn_wmma_f32_16x16x4_f32(false, a1o, false, bo1, (short)0, af1, false, false);
    af2 = __builtin_amdgcn_wmma_f32_16x16x4_f32(false, a1o, false, bo2, (short)0, af2, false, false);
    af3 = __builtin_amdgcn_wmma_f32_16x16x4_f32(false, a1o, false, bo3, (short)0, af3, false, false);
  }

  // D layout: acc[r] -> row (rowbase + r + 8*half), col (nt*16 + t).
#pragma unroll
  for (int rt = 0; rt < 2; ++rt) {
    const int rbase = row0 + rt * 16;
    float rs[8];
#pragma unroll
    for (int r = 0; r < 8; ++r) {
      const int rr = rbase + r + 8 * half;
      rs[r] = rsqrtf((float)(colptr[rr + 1] - colptr[rr]));
    }
#pragma unroll
    for (int r = 0; r < 8; ++r) {
      const int rr = rbase + r + 8 * half;
      float* hp = h + (size_t)rr * DOUT + t;
      if (rt == 0) {
        hp[0] = ae0[r] * rs[r]; hp[16] = ae1[r] * rs[r];
        hp[32] = ae2[r] * rs[r]; hp[48] = ae3[r] * rs[r];
      } else {
        hp[0] = af0[r] * rs[r]; hp[16] = af1[r] * rs[r];
        hp[32] = af2[r] * rs[r]; hp[48] = af3[r] * rs[r];
      }
    }
  }
}

// out[i, :] = rsqrt(in_deg[i]) * sum_e h[colind[e], :] + bias
// One thread per (row, 4-col quad); h gathers stay L2-resident (25.6 MB << 192 MB).
__global__ __launch_bounds__(256) void aggr_kernel(
    const float* __restrict__ h, const int* __restrict__ rowptr,
    const int* __restrict__ colind, const float* __restrict__ bias,
    float* __restrict__ out, int n) {
  const int idx = blockIdx.x * 256 + threadIdx.x;
  const int row = idx >> 4;
  const int q   = (idx & 15) * 4;
  if (row >= n) return;

  const int s = rowptr[row];
  const int e = rowptr[row + 1];

  v4f acc = {};
#pragma unroll 4
  for (int j = s; j < e; ++j) {
    const int c = colind[j];
    v4f v = *(const v4f*)(h + (size_t)c * DOUT + q);
    acc += v;
  }

  const float sc = rsqrtf((float)(e - s));
  v4f b4 = *(const v4f*)(bias + q);
  v4f r = acc * sc + b4;
  __builtin_nontemporal_store(r, (v4f*)(out + (size_t)row * DOUT + q));
}

extern "C" void kernel_launch(void* const* d_in, const int* in_sizes, int n_in,
                              void* d_out, int out_size, void* d_ws, size_t ws_size,
                              hipStream_t stream) {
  const float* x      = (const float*)d_in[0];
  const float* w      = (const float*)d_in[1];
  const float* bias   = (const float*)d_in[2];
  const int*   rowptr = (const int*)d_in[3];
  const int*   colind = (const int*)d_in[4];
  const int*   colptr = (const int*)d_in[5];
  // d_in[6] (rowind) unused: only degrees are needed from the CSC side.

  const int n = in_sizes[3] - 1;  // rowptr has N+1 entries
  float* h = (float*)d_ws;        // [n, DOUT] scratch = 25.6 MB

  const int tiles = (n + ROWS_PER_WAVE - 1) / ROWS_PER_WAVE;
  dim3 g1((tiles + WAVES_PER_BLOCK - 1) / WAVES_PER_BLOCK);
  gemm_scale_kernel<<<g1, 256, 0, stream>>>(x, w, colptr, h, n);

  const int total = n * 16;
  dim3 g2((total + 255) / 256);
  aggr_kernel<<<g2, 256, 0, stream>>>(h, rowptr, colind, bias, (float*)d_out, n);
}